// EEM_34488587386947
// MI455X (gfx1250) — compile-verified
//
#include <hip/hip_runtime.h>
#include <math.h>

// ---------------------------------------------------------------------------
// MI455X (gfx1250) implementation.
// B=8, C_in=2, H=W=320, C2=48, HEADS=8, c=6 per head.
//
// Pass 1: per 16x16 pixel tile, fused LN + 1x1 conv + 3x3 dw conv -> q,
//         3x3 conv(e) -> k, staged in LDS; then each wave (== one head)
//         accumulates the 6x6 Gram q*k^T plus ||q||^2/||k||^2 diagonals
//         with V_WMMA_F32_16X16X4_F32 (3 wmma per K=4 step).
//         Lanes with lanelo>=6 carry clamped (finite) data; they only feed
//         C rows/cols >= 6 which are never extracted, so no masking needed.
// Pass 2: tiny per-(b,h) kernel: normalize Gram, softmax, fold attn with
//         wproj into per-batch 2x48 matrix M.
// Pass 3: stream x again, recompute v (2nd half of conv chain) per pixel,
//         out = x + M @ v.
// ---------------------------------------------------------------------------

#define BB 8
#define HH 320
#define WW 320
#define HWTOT (HH * WW)
#define C2 48
#define HEADS 8
#define CPH 6              // channels per head
#define TILE 16
#define TPIX (TILE * TILE) // 256 pixels per tile

typedef __attribute__((ext_vector_type(2))) float v2f;
typedef __attribute__((ext_vector_type(8))) float v8f;

// Workspace layout (floats):
//   G  : B*8*6*6 = 2304   raw Gram accumulators
//   q2 : B*48    =  384   ||q_c||^2
//   k2 : B*48    =  384   ||k_d||^2
//   M  : B*2*48  =  768   folded attn*wproj
#define WS_G 0
#define WS_Q2 2304
#define WS_K2 2688
#define WS_M 3072
#define WS_ZERO_FLOATS 3072

// ---------------------------------------------------------------------------
__global__ void eem_pass1(const float* __restrict__ x, const float* __restrict__ e,
                          const float* __restrict__ ln_w, const float* __restrict__ w1,
                          const float* __restrict__ w2, const float* __restrict__ we,
                          float* __restrict__ G, float* __restrict__ q2,
                          float* __restrict__ k2) {
  extern __shared__ float smem[];
  float* t0s = smem;          // 18*18 LN'd channel 0 (with halo)
  float* t1s = smem + 324;    // 18*18 LN'd channel 1
  float* es  = smem + 648;    // 18*18 edge map
  float* qt  = smem + 972;    // [48][256] q tile
  float* kt  = qt + C2 * TPIX; // [48][256] k tile

  const int b = blockIdx.z;
  const int bx = blockIdx.x * TILE, by = blockIdx.y * TILE;
  const int t = threadIdx.x;
  const float lw0 = ln_w[0], lw1 = ln_w[1];
  const float* xb = x + (size_t)b * 2 * HWTOT;
  const float* eb = e + (size_t)b * HWTOT;

  // ---- stage 1: halo load + BiasFree LayerNorm (2-channel closed form) ----
  for (int i = t; i < 18 * 18; i += TPIX) {
    int hy = i / 18, hx = i % 18;
    int gy = by + hy - 1, gx = bx + hx - 1;
    float v0 = 0.f, v1 = 0.f, ev = 0.f;
    if (gy >= 0 && gy < HH && gx >= 0 && gx < WW) {
      float x0 = xb[gy * WW + gx];
      float x1 = xb[HWTOT + gy * WW + gx];
      float d = x0 - x1;                     // var over 2 ch = ((x0-x1)/2)^2
      float r = rsqrtf(0.25f * d * d + 1e-5f);
      v0 = x0 * r * lw0;
      v1 = x1 * r * lw1;
      ev = eb[gy * WW + gx];
    }
    t0s[i] = v0; t1s[i] = v1; es[i] = ev;
  }
  __syncthreads();

  // ---- stage 2: per-pixel q (48ch) and k (48ch) into LDS tiles ----
  {
    int ly = t >> 4, lx = t & 15;
    float n0[9], n1[9], ne[9];
#pragma unroll
    for (int ky = 0; ky < 3; ky++)
#pragma unroll
      for (int kx = 0; kx < 3; kx++) {
        int idx = (ly + ky) * 18 + (lx + kx);
        n0[ky * 3 + kx] = t0s[idx];
        n1[ky * 3 + kx] = t1s[idx];
        ne[ky * 3 + kx] = es[idx];
      }
    for (int oc = 0; oc < C2; oc++) {
      float a0 = 0.f, a1v = 0.f, kv = 0.f;
#pragma unroll
      for (int j = 0; j < 9; j++) {
        float w = w2[oc * 9 + j];          // dw-conv weight, q half
        a0 += w * n0[j];
        a1v += w * n1[j];
        kv += we[oc * 9 + j] * ne[j];      // edge-head conv
      }
      // q[oc] = w1[oc,0]*(w2 conv t0) + w1[oc,1]*(w2 conv t1)
      qt[oc * TPIX + t] = w1[oc * 2 + 0] * a0 + w1[oc * 2 + 1] * a1v;
      kt[oc * TPIX + t] = kv;
    }
  }
  __syncthreads();

  // ---- stage 3: WMMA Gram accumulation, one wave per head ----
  const int h = t >> 5;           // wave id == head
  const int lane = t & 31;
  const int lanelo = lane & 15;
  const int hi = lane >> 4;       // K-half selector per ISA A/B layout
  // Clamp invalid rows to a real channel: finite data that only reaches
  // C rows/cols >= 6, which are never extracted below.
  const int row = h * CPH + (lanelo < CPH ? lanelo : 0);
  const v2f* qv = (const v2f*)(qt + row * TPIX); // 8B-aligned pairs
  const v2f* kv = (const v2f*)(kt + row * TPIX);

  v8f cqk = {}; // q * k^T   (Gram)
  v8f cqq = {}; // q * q^T   (need diag only)
  v8f ckk = {}; // k * k^T   (need diag only)
#pragma unroll 4
  for (int i = 0; i < TPIX / 4; ++i) {
    // A lane layout: (M=lanelo, K=hi*2+j) ; B: (K=hi*2+j, N=lanelo).
    // Symmetric Gram => A- and B-fragment layouts coincide.
    v2f aq = qv[i * 2 + hi]; // one ds_load_b64
    v2f bk = kv[i * 2 + hi]; // one ds_load_b64
    cqk = __builtin_amdgcn_wmma_f32_16x16x4_f32(false, aq, false, bk, (short)0, cqk, false, false);
    cqq = __builtin_amdgcn_wmma_f32_16x16x4_f32(false, aq, false, aq, (short)0, cqq, false, false);
    ckk = __builtin_amdgcn_wmma_f32_16x16x4_f32(false, bk, false, bk, (short)0, ckk, false, false);
  }

  // C layout: VGPR r, lanes 0-15 -> (M=r, N=lanelo); lanes 16-31 -> (M=r+8).
  if (hi == 0 && lanelo < CPH) {
#pragma unroll
    for (int r = 0; r < CPH; r++) {
      atomicAdd(&G[((b * HEADS + h) * CPH + r) * CPH + lanelo], cqk[r]);
      if (lanelo == r) {
        atomicAdd(&q2[b * C2 + h * CPH + r], cqq[r]);
        atomicAdd(&k2[b * C2 + h * CPH + r], ckk[r]);
      }
    }
  }
}

// ---------------------------------------------------------------------------
__global__ void eem_pass2(const float* __restrict__ G, const float* __restrict__ q2,
                          const float* __restrict__ k2, const float* __restrict__ a1,
                          const float* __restrict__ wproj, float* __restrict__ M) {
  const int b = blockIdx.x / HEADS;
  const int h = blockIdx.x % HEADS;
  const int t = threadIdx.x; // 64 threads
  __shared__ float s[CPH * CPH];
  __shared__ float attn[CPH * CPH];

  if (t < CPH * CPH) {
    int c = t / CPH, d = t % CPH;
    float qn = fmaxf(sqrtf(q2[b * C2 + h * CPH + c]), 1e-12f);
    float kn = fmaxf(sqrtf(k2[b * C2 + h * CPH + d]), 1e-12f);
    s[t] = G[((b * HEADS + h) * CPH + c) * CPH + d] / (qn * kn) * a1[h];
  }
  __syncthreads();
  if (t < CPH * CPH) {
    int c = t / CPH;
    float mx = -INFINITY;
#pragma unroll
    for (int d = 0; d < CPH; d++) mx = fmaxf(mx, s[c * CPH + d]);
    float sum = 0.f;
#pragma unroll
    for (int d = 0; d < CPH; d++) sum += __expf(s[c * CPH + d] - mx);
    attn[t] = __expf(s[t] - mx) / sum;
  }
  __syncthreads();
  // Fold projection: M[o][h*6+d] = sum_c wproj[o][h*6+c] * attn[c][d]
  if (t < 2 * CPH) {
    int o = t / CPH, d = t % CPH;
    float acc = 0.f;
#pragma unroll
    for (int c = 0; c < CPH; c++)
      acc += wproj[o * C2 + h * CPH + c] * attn[c * CPH + d];
    M[(b * 2 + o) * C2 + h * CPH + d] = acc;
  }
}

// ---------------------------------------------------------------------------
__global__ void eem_pass3(const float* __restrict__ x, const float* __restrict__ ln_w,
                          const float* __restrict__ w1, const float* __restrict__ w2,
                          const float* __restrict__ M, float* __restrict__ out) {
  extern __shared__ float smem[];
  float* t0s = smem;       // 18*18
  float* t1s = smem + 324; // 18*18

  const int b = blockIdx.z;
  const int bx = blockIdx.x * TILE, by = blockIdx.y * TILE;
  const int t = threadIdx.x;
  const float lw0 = ln_w[0], lw1 = ln_w[1];
  const float* xb = x + (size_t)b * 2 * HWTOT;

  for (int i = t; i < 18 * 18; i += TPIX) {
    int hy = i / 18, hx = i % 18;
    int gy = by + hy - 1, gx = bx + hx - 1;
    float v0 = 0.f, v1 = 0.f;
    if (gy >= 0 && gy < HH && gx >= 0 && gx < WW) {
      float x0 = xb[gy * WW + gx];
      float x1 = xb[HWTOT + gy * WW + gx];
      float d = x0 - x1;
      float r = rsqrtf(0.25f * d * d + 1e-5f);
      v0 = x0 * r * lw0;
      v1 = x1 * r * lw1;
    }
    t0s[i] = v0; t1s[i] = v1;
  }
  __syncthreads();

  int ly = t >> 4, lx = t & 15;
  float n0[9], n1[9];
#pragma unroll
  for (int ky = 0; ky < 3; ky++)
#pragma unroll
    for (int kx = 0; kx < 3; kx++) {
      int idx = (ly + ky) * 18 + (lx + kx);
      n0[ky * 3 + kx] = t0s[idx];
      n1[ky * 3 + kx] = t1s[idx];
    }

  const float* M0 = M + (size_t)b * 2 * C2;
  const float* M1 = M0 + C2;
  float acc0 = 0.f, acc1 = 0.f;
  for (int j = 0; j < C2; j++) {
    int oc = C2 + j; // v = second half of the conv chain
    float a0 = 0.f, a1v = 0.f;
#pragma unroll
    for (int k = 0; k < 9; k++) {
      float w = w2[oc * 9 + k];
      a0 += w * n0[k];
      a1v += w * n1[k];
    }
    float vv = w1[oc * 2 + 0] * a0 + w1[oc * 2 + 1] * a1v;
    acc0 += M0[j] * vv;
    acc1 += M1[j] * vv;
  }
  int gy = by + ly, gx = bx + lx;
  out[((size_t)b * 2 + 0) * HWTOT + gy * WW + gx] = xb[gy * WW + gx] + acc0;
  out[((size_t)b * 2 + 1) * HWTOT + gy * WW + gx] = xb[HWTOT + gy * WW + gx] + acc1;
}

// ---------------------------------------------------------------------------
extern "C" void kernel_launch(void* const* d_in, const int* in_sizes, int n_in,
                              void* d_out, int out_size, void* d_ws, size_t ws_size,
                              hipStream_t stream) {
  const float* x     = (const float*)d_in[0];
  const float* e     = (const float*)d_in[1];
  const float* ln_w  = (const float*)d_in[2];
  const float* w1    = (const float*)d_in[3];
  const float* w2    = (const float*)d_in[4];
  const float* we    = (const float*)d_in[5];
  const float* a1    = (const float*)d_in[6];
  const float* wproj = (const float*)d_in[7];
  float* out = (float*)d_out;

  float* ws = (float*)d_ws;
  float* G  = ws + WS_G;
  float* q2 = ws + WS_Q2;
  float* k2 = ws + WS_K2;
  float* M  = ws + WS_M;

  // zero the Gram / norm accumulators every launch (graph-capturable)
  hipMemsetAsync(ws, 0, WS_ZERO_FLOATS * sizeof(float), stream);

  dim3 grid(WW / TILE, HH / TILE, BB); // 20 x 20 x 8 tiles
  dim3 blk(TPIX);

  size_t sh1 = (size_t)(3 * 324 + 2 * C2 * TPIX) * sizeof(float); // ~100 KB LDS
  eem_pass1<<<grid, blk, sh1, stream>>>(x, e, ln_w, w1, w2, we, G, q2, k2);

  eem_pass2<<<BB * HEADS, 64, 0, stream>>>(G, q2, k2, a1, wproj, M);

  size_t sh3 = (size_t)(2 * 324) * sizeof(float);
  eem_pass3<<<grid, blk, sh3, stream>>>(x, ln_w, w1, w2, M, out);
}